// MoELatent_48404281426530
// MI455X (gfx1250) — compile-verified
//
#include <hip/hip_runtime.h>
#include <hip/hip_bf16.h>

// ---------------- problem constants (match reference) ----------------
static constexpr int BT = 2048;   // tokens
static constexpr int DD = 1024;   // model dim
static constexpr int HH = 2048;   // hidden dim
static constexpr int EE = 16;     // experts
static constexpr int WW = 4;      // adapters per expert
static constexpr int RR = 8;      // LoRA rank
static constexpr int KK = 2;      // top_k (reference: min(2, E))

// d_out layout (floats), concatenated in reference return order
static constexpr size_t O_Z     = 0;                                   // [B,D]
static constexpr size_t O_PROBS = O_Z     + (size_t)BT * DD;           // [B,E]
static constexpr size_t O_MASK  = O_PROBS + (size_t)BT * EE;           // [B,E]
static constexpr size_t O_AIDX  = O_MASK  + (size_t)BT * EE;           // [B,k]
static constexpr size_t O_AW    = O_AIDX  + (size_t)BT * KK;           // [B,k]
static constexpr size_t O_HIT   = O_AW    + (size_t)BT * KK;           // scalar
static constexpr size_t O_EVE   = O_HIT   + 1;                         // [k*B]
static constexpr size_t O_EVA   = O_EVE   + (size_t)BT * KK;           // [k*B]

typedef __attribute__((ext_vector_type(2))) float v2f;
typedef __attribute__((ext_vector_type(8))) float v8f;

__device__ __forceinline__ float silu_f(float x) {
    return x / (1.0f + __expf(-x));
}

// CDNA5 async copy: global -> LDS, 16B per lane, tracked by ASYNCcnt.
// VDST = 32-bit LDS byte address, VADDR = 64-bit global address.
__device__ __forceinline__ void async_b128_to_lds(const float* gaddr, unsigned lds_addr) {
    asm volatile("global_load_async_to_lds_b128 %0, %1, off"
                 :: "v"(lds_addr), "v"((unsigned long long)(uintptr_t)gaddr)
                 : "memory");
}

// =====================================================================
// Kernel 1/2: WMMA f32 GEMM  C[M,N] = act(A[M,K] @ Bw[N,K]^T + bias[N])
// 128x64 block tile, 8 waves in 4(M)x2(N) grid, 4 accumulators per wave
// (32x32 output each; A/B fragments each reused twice -> 4 WMMAs per 4
// LDS fragment loads). K staged through double-buffered LDS (stride-20
// pad) via GLOBAL_LOAD_ASYNC_TO_LDS_B128: tile k+1 streams while tile k
// computes; per-tile 3 async ops -> s_wait_asynccnt 3 overlap.
// =====================================================================
__global__ __launch_bounds__(256)
void wmma_gemm_bias_act(const float* __restrict__ A, const float* __restrict__ Bw,
                        const float* __restrict__ bias, float* __restrict__ C,
                        int M, int N, int K, int apply_silu) {
    __shared__ float As[2][128 * 20];   // 2 x 10 KB
    __shared__ float Bs[2][64 * 20];    // 2 x  5 KB

    const int tid  = threadIdx.x;
    const int lane = tid & 31;
    const int wave = tid >> 5;
    const int wm   = wave & 3;           // 0..3  (M sub-block of 32)
    const int wn   = wave >> 2;          // 0..1  (N sub-block of 32)

    const int m0 = blockIdx.y * 128;
    const int n0 = blockIdx.x * 64;

    // ---- tile loader mapping: 16B chunks ----
    // A tile: 128 rows x 16 cols = 512 chunks, 2 per thread
    const int ca0 = tid;                 // chunk 0
    const int ca1 = tid + 256;           // chunk 1
    const int ra0 = ca0 >> 2, cc0 = (ca0 & 3) << 2;
    const int ra1 = ca1 >> 2, cc1 = (ca1 & 3) << 2;
    // B tile: 64 rows x 16 cols = 256 chunks, 1 per thread
    const int rb  = tid >> 2,  ccb = (tid & 3) << 2;

    const float* gA0 = A  + (size_t)(m0 + ra0) * K + cc0;
    const float* gA1 = A  + (size_t)(m0 + ra1) * K + cc1;
    const float* gB0 = Bw + (size_t)(n0 + rb)  * K + ccb;

    const unsigned lA0[2] = { (unsigned)(uintptr_t)&As[0][ra0 * 20 + cc0],
                              (unsigned)(uintptr_t)&As[1][ra0 * 20 + cc0] };
    const unsigned lA1[2] = { (unsigned)(uintptr_t)&As[0][ra1 * 20 + cc1],
                              (unsigned)(uintptr_t)&As[1][ra1 * 20 + cc1] };
    const unsigned lB0[2] = { (unsigned)(uintptr_t)&Bs[0][rb * 20 + ccb],
                              (unsigned)(uintptr_t)&Bs[1][rb * 20 + ccb] };

    const int mrow = lane & 15;          // row within 16
    const int kh   = (lane >> 4) << 1;   // 0 or 2 (K-half per ISA layout)

    v8f acc00 = {}, acc01 = {}, acc10 = {}, acc11 = {};

    const int ktiles = K >> 4;

    // prologue: stream tile 0 into buffer 0
    async_b128_to_lds(gA0, lA0[0]);
    async_b128_to_lds(gA1, lA1[0]);
    async_b128_to_lds(gB0, lB0[0]);

    for (int kt = 0; kt < ktiles; ++kt) {
        const int cur = kt & 1;

        // stream next tile into the other buffer, then wait for current:
        // async loads complete in order, so asynccnt<=3 (the next tile's
        // three in-flight ops) implies the current tile has landed.
        if (kt + 1 < ktiles) {
            const int koff = (kt + 1) << 4;
            const int nb = cur ^ 1;
            async_b128_to_lds(gA0 + koff, lA0[nb]);
            async_b128_to_lds(gA1 + koff, lA1[nb]);
            async_b128_to_lds(gB0 + koff, lB0[nb]);
            asm volatile("s_wait_asynccnt 0x3" ::: "memory");
        } else {
            asm volatile("s_wait_asynccnt 0x0" ::: "memory");
        }
        __syncthreads();   // current tile visible to all waves

        // ---- 4 K-substeps of 4, 4 WMMAs each (A,B frags reused 2x) ----
        #pragma unroll
        for (int kk = 0; kk < 16; kk += 4) {
            const int abase = (wm * 32 + mrow) * 20 + kk + kh;
            const int bbase = (wn * 32 + mrow) * 20 + kk + kh;

            v2f af0; af0.x = As[cur][abase];            af0.y = As[cur][abase + 1];
            v2f af1; af1.x = As[cur][abase + 16 * 20];  af1.y = As[cur][abase + 16 * 20 + 1];
            v2f bf0; bf0.x = Bs[cur][bbase];            bf0.y = Bs[cur][bbase + 1];
            v2f bf1; bf1.x = Bs[cur][bbase + 16 * 20];  bf1.y = Bs[cur][bbase + 16 * 20 + 1];

            acc00 = __builtin_amdgcn_wmma_f32_16x16x4_f32(
                false, af0, false, bf0, (short)0, acc00, false, false);
            acc01 = __builtin_amdgcn_wmma_f32_16x16x4_f32(
                false, af0, false, bf1, (short)0, acc01, false, false);
            acc10 = __builtin_amdgcn_wmma_f32_16x16x4_f32(
                false, af1, false, bf0, (short)0, acc10, false, false);
            acc11 = __builtin_amdgcn_wmma_f32_16x16x4_f32(
                false, af1, false, bf1, (short)0, acc11, false, false);
        }
        __syncthreads();   // everyone done reading before buffer re-streams
    }

    // ---- epilogue: C/D layout => VGPR v holds row v (lanes 0-15) / v+8 (16-31)
    const int mb     = m0 + wm * 32;
    const int nb     = n0 + wn * 32;
    const int ncol   = lane & 15;
    const int rowoff = (lane >> 4) * 8;
    const float bv0  = bias[nb + ncol];
    const float bv1  = bias[nb + 16 + ncol];
    #pragma unroll
    for (int v = 0; v < 8; ++v) {
        const int r0 = mb + rowoff + v;
        const int r1 = mb + 16 + rowoff + v;
        float x00 = acc00[v] + bv0, x01 = acc01[v] + bv1;
        float x10 = acc10[v] + bv0, x11 = acc11[v] + bv1;
        if (apply_silu) {
            x00 = silu_f(x00); x01 = silu_f(x01);
            x10 = silu_f(x10); x11 = silu_f(x11);
        }
        C[(size_t)r0 * N + nb + ncol]      = x00;
        C[(size_t)r0 * N + nb + 16 + ncol] = x01;
        C[(size_t)r1 * N + nb + ncol]      = x10;
        C[(size_t)r1 * N + nb + 16 + ncol] = x11;
    }
}

// =====================================================================
// Kernel 3: router probabilities (one wave per token) + zero used_mask
// =====================================================================
__global__ __launch_bounds__(256)
void router_probs(const float* __restrict__ z, const float* __restrict__ proto,
                  const float* __restrict__ ebias,
                  const float* __restrict__ tau_p, const float* __restrict__ eps_p,
                  const int* __restrict__ ban_p,
                  float* __restrict__ probs_out, float* __restrict__ mask_out) {
    const int b    = blockIdx.x * 8 + (threadIdx.x >> 5);
    const int lane = threadIdx.x & 31;
    if (b >= BT) return;

    const float tau = fmaxf(tau_p[0], 1e-6f);
    const float eps = eps_p[0];
    const int   ban = ban_p[0];
    const float* zr = z + (size_t)b * DD;

    float logit[EE];
    for (int e = 0; e < EE; ++e) {
        const float* pr = proto + (size_t)e * DD;
        float s = 0.0f;
        for (int d = lane; d < DD; d += 32) s += zr[d] * pr[d];
        for (int off = 16; off; off >>= 1) s += __shfl_xor(s, off, 32);
        float l = s / tau + ebias[e];
        if (e == ban) l = -1e9f;
        logit[e] = l;
    }
    // redundant per-lane softmax over 16 (cheap)
    float mx = logit[0];
    for (int e = 1; e < EE; ++e) mx = fmaxf(mx, logit[e]);
    float den = 0.0f;
    for (int e = 0; e < EE; ++e) den += __expf(logit[e] - mx);
    if (lane < EE) {
        float p = __expf(logit[lane] - mx) / den;
        p = (1.0f - eps) * p + eps / (float)EE;
        probs_out[(size_t)b * EE + lane] = p;
        mask_out[(size_t)b * EE + lane]  = 0.0f;   // zero-init used_mask
    }
}

// =====================================================================
// Kernel 4: sequential greedy capacity routing (order-dependent scan).
// The token loop must stay serial (shared capacity counters), but:
//  - probs stream into LDS in 32KB chunks via async-to-LDS, double
//    buffered: waves 1-7 prefetch chunk c+1 while wave 0 scans chunk c.
//  - wave 0 scans cooperatively: lanes 0-15 hold the 16 expert probs,
//    top-8 selection = 8 rounds of shfl_xor argmax (first-occurrence
//    tie-break), capacity state = 1 register per lane (lane e = cap[e])
//    queried with a dynamic shuffle. No global latency on critical path.
// =====================================================================
static constexpr int CH = 512;     // tokens per staged chunk (32 KB)

__global__ __launch_bounds__(256)
void routing_scan(const float* __restrict__ probs,
                  const int* __restrict__ topk_p,
                  const int* __restrict__ cap_p,
                  float* __restrict__ out_aidx, float* __restrict__ out_aw,
                  float* __restrict__ out_mask, float* __restrict__ out_eve,
                  float* __restrict__ out_hit,
                  int* __restrict__ ws_e, float* __restrict__ ws_w) {
    __shared__ float pb[2][CH * EE];   // 2 x 32 KB

    const int tid  = threadIdx.x;
    const int lane = tid & 31;
    const int wave = tid >> 5;

    int k = topk_p[0];
    if (k > KK) k = KK;
    if (k < 1)  k = 1;
    int cand_k = 4 * k; if (cand_k > EE) cand_k = EE;
    int cap_limit = cap_p[0]; if (cap_limit < 1) cap_limit = 1;

    // prologue: stage chunk 0 (all threads)
    for (int i = tid; i < CH * EE / 4; i += 256)
        async_b128_to_lds(probs + i * 4, (unsigned)(uintptr_t)&pb[0][i * 4]);
    asm volatile("s_wait_asynccnt 0x0" ::: "memory");
    __syncthreads();

    int capreg = 0;   // lane e (<16): capacity count of expert e
    int hits   = 0;
    const int nch = BT / CH;

    for (int c = 0; c < nch; ++c) {
        const int cur = c & 1;

        // waves 1-7: prefetch next chunk into the other buffer
        if (wave > 0 && c + 1 < nch) {
            const float* src = probs + (size_t)(c + 1) * CH * EE;
            float* dst = &pb[cur ^ 1][0];
            for (int i = tid - 32; i < CH * EE / 4; i += 224)
                async_b128_to_lds(src + i * 4, (unsigned)(uintptr_t)&dst[i * 4]);
            asm volatile("s_wait_asynccnt 0x0" ::: "memory");
        }

        // wave 0: scan this chunk serially over tokens
        if (wave == 0) {
            for (int t = 0; t < CH; ++t) {
                const int b = c * CH + t;
                float v = (lane < EE) ? pb[cur][t * EE + lane] : -1e30f;

                int   cid[8];
                float cv[8];
                #pragma unroll
                for (int cc = 0; cc < 8; ++cc) {
                    float bv = v;
                    int   bi = lane;
                    #pragma unroll
                    for (int off = 1; off < 16; off <<= 1) {
                        const float ov = __shfl_xor(bv, off, 16);
                        const int   oi = __shfl_xor(bi, off, 16);
                        if (ov > bv || (ov == bv && oi < bi)) { bv = ov; bi = oi; }
                    }
                    cid[cc] = bi;
                    cv[cc]  = bv;
                    if (lane == bi) v = -1e30f;
                }

                int   a0 = -1, a1 = -1;
                float w0 = 0.0f, w1 = 0.0f;
                int   slot = 0;
                #pragma unroll
                for (int j = 0; j < 8; ++j) {
                    if (j < cand_k) {
                        const int  e    = cid[j];
                        const int  cape = __shfl(capreg, e, 16);
                        const bool take = (slot < k) && (cape < cap_limit);
                        if (take) {
                            if (slot == 0) { a0 = e; w0 = cv[j]; }
                            else           { a1 = e; w1 = cv[j]; }
                            if (lane == e) capreg++;
                            if (j > 0) hits++;
                            slot++;
                        }
                    }
                }

                if (lane == 0) {
                    out_aidx[(size_t)b * KK + 0] = (float)a0;
                    out_aidx[(size_t)b * KK + 1] = (float)a1;
                    out_aw[(size_t)b * KK + 0]   = w0;
                    out_aw[(size_t)b * KK + 1]   = w1;
                    out_eve[(size_t)0 * BT + b]  = (float)a0;
                    out_eve[(size_t)1 * BT + b]  = (float)a1;
                    ws_e[b * KK + 0] = a0;
                    ws_e[b * KK + 1] = a1;
                    ws_w[b * KK + 0] = w0;
                    ws_w[b * KK + 1] = w1;
                    if (a0 >= 0) out_mask[(size_t)b * EE + a0] = 1.0f;
                    if (a1 >= 0) out_mask[(size_t)b * EE + a1] = 1.0f;
                }
            }
        }
        __syncthreads();
    }
    if (tid == 0) out_hit[0] = (float)hits / (float)(BT * k);
}

// =====================================================================
// Kernel 5: adapter selection, one wave per (token, slot)
// =====================================================================
__global__ __launch_bounds__(256)
void adapter_argmax(const float* __restrict__ z, const float* __restrict__ akeys,
                    const int* __restrict__ ws_e, int* __restrict__ ws_a,
                    float* __restrict__ out_eva) {
    const int slot = blockIdx.x * 8 + (threadIdx.x >> 5);
    const int lane = threadIdx.x & 31;
    if (slot >= BT * KK) return;
    const int b = slot >> 1;
    const int s = slot & 1;

    const int e     = ws_e[slot];
    const int esafe = (e >= 0) ? e : 0;
    const float* zr = z + (size_t)b * DD;

    int   bw   = 0;
    float best = -1e30f;
    for (int w = 0; w < WW; ++w) {
        const float* kr = akeys + ((size_t)esafe * WW + w) * DD;
        float acc = 0.0f;
        for (int d = lane; d < DD; d += 32) acc += zr[d] * kr[d];
        for (int off = 16; off; off >>= 1) acc += __shfl_xor(acc, off, 32);
        if (acc > best) { best = acc; bw = w; }   // first-max tie break
    }
    if (lane == 0) {
        ws_a[slot] = bw;
        out_eva[(size_t)s * BT + b] = (e >= 0) ? (float)bw : -1.0f;
    }
}

// =====================================================================
// Kernel 6: LoRA adapters + weighted combine, one block per token.
//  A: t1[s][r] = z . A1[r,:]          (wave-parallel rank dots)
//  B: d1 -> silu -> t2 partials       (H-parallel, LDS ds_add_f32 reduce)
//  C: d2 + w*(out_pre+d2) combine     (D-parallel)
// =====================================================================
__global__ __launch_bounds__(256)
void lora_combine(const float* __restrict__ z, const float* __restrict__ h_pre,
                  const float* __restrict__ out_pre,
                  const float* __restrict__ a1, const float* __restrict__ b1,
                  const float* __restrict__ a2, const float* __restrict__ b2,
                  const int* __restrict__ ws_e, const int* __restrict__ ws_a,
                  const float* __restrict__ ws_w, float* __restrict__ out) {
    const int b    = blockIdx.x;
    const int tid  = threadIdx.x;
    const int lane = tid & 31;
    const int wave = tid >> 5;

    __shared__ float t1s[KK][RR];
    __shared__ float t2s[KK][RR];
    __shared__ int   sh_e[KK];
    __shared__ int   sh_a[KK];
    __shared__ float sh_w[KK];

    if (tid < KK) {
        sh_e[tid] = ws_e[b * KK + tid];
        sh_a[tid] = ws_a[b * KK + tid];
        sh_w[tid] = ws_w[b * KK + tid];
    }
    if (tid < KK * RR) t2s[tid >> 3][tid & 7] = 0.0f;
    __syncthreads();

    // ---- stage A: wave w computes t1[s][w] for s = 0,1 ----
    const float* zr = z + (size_t)b * DD;
    for (int s = 0; s < KK; ++s) {
        const int e = sh_e[s];
        float acc = 0.0f;
        if (e >= 0) {
            const float* A1r = a1 + (((size_t)(e * WW + sh_a[s]) * RR + wave)) * DD;
            for (int d = lane; d < DD; d += 32) acc += zr[d] * A1r[d];
        }
        for (int off = 16; off; off >>= 1) acc += __shfl_xor(acc, off, 32);
        if (lane == 0) t1s[s][wave] = acc;
    }
    __syncthreads();

    // ---- stage B: H-parallel d1 -> silu -> t2 partials ----
    float t2loc[KK][RR];
    #pragma unroll
    for (int s = 0; s < KK; ++s)
        #pragma unroll
        for (int r = 0; r < RR; ++r) t2loc[s][r] = 0.0f;

    for (int h = tid; h < HH; h += 256) {
        const float hp = h_pre[(size_t)b * HH + h];
        #pragma unroll
        for (int s = 0; s < KK; ++s) {
            const int e = sh_e[s];
            if (e < 0) continue;
            const size_t base = (size_t)(e * WW + sh_a[s]);
            const float* B1r = b1 + (base * HH + h) * RR;    // 8 contiguous
            float d1 = 0.0f;
            #pragma unroll
            for (int r = 0; r < RR; ++r) d1 += t1s[s][r] * B1r[r];
            const float hv = silu_f(hp + d1);
            const float* A2b = a2 + base * RR * HH;
            #pragma unroll
            for (int r = 0; r < RR; ++r) t2loc[s][r] += hv * A2b[(size_t)r * HH + h];
        }
    }
    #pragma unroll
    for (int s = 0; s < KK; ++s) {
        if (sh_e[s] < 0) continue;
        #pragma unroll
        for (int r = 0; r < RR; ++r) atomicAdd(&t2s[s][r], t2loc[s][r]);
    }
    __syncthreads();

    // ---- stage C: D-parallel d2 + combine ----
    #pragma unroll
    for (int i = 0; i < DD / 256; ++i) {
        const int d = tid + i * 256;
        float v  = zr[d];
        const float op = out_pre[(size_t)b * DD + d];
        #pragma unroll
        for (int s = 0; s < KK; ++s) {
            const int e = sh_e[s];
            if (e < 0) continue;
            const float* B2r = b2 + (((size_t)(e * WW + sh_a[s]) * DD + d)) * RR;
            float d2 = 0.0f;
            #pragma unroll
            for (int r = 0; r < RR; ++r) d2 += t2s[s][r] * B2r[r];
            v += sh_w[s] * (op + d2);
        }
        out[(size_t)b * DD + d] = v;
    }
}

// =====================================================================
extern "C" void kernel_launch(void* const* d_in, const int* in_sizes, int n_in,
                              void* d_out, int out_size, void* d_ws, size_t ws_size,
                              hipStream_t stream) {
    const float* z      = (const float*)d_in[0];
    const int*   top_k  = (const int*)  d_in[1];
    const int*   cap    = (const int*)  d_in[2];
    const int*   ban    = (const int*)  d_in[3];
    const float* tau    = (const float*)d_in[4];
    const float* eps    = (const float*)d_in[5];
    const float* fc1_w  = (const float*)d_in[6];
    const float* fc1_b  = (const float*)d_in[7];
    const float* fc2_w  = (const float*)d_in[8];
    const float* fc2_b  = (const float*)d_in[9];
    const float* proto  = (const float*)d_in[10];
    const float* akeys  = (const float*)d_in[11];
    const float* ebias  = (const float*)d_in[12];
    const float* a1     = (const float*)d_in[13];
    const float* b1     = (const float*)d_in[14];
    const float* a2     = (const float*)d_in[15];
    const float* b2     = (const float*)d_in[16];

    float* out = (float*)d_out;

    // workspace carve-up
    float* h_pre   = (float*)d_ws;                          // [B,H]
    float* out_pre = h_pre + (size_t)BT * HH;               // [B,D]
    int*   ws_e    = (int*)(out_pre + (size_t)BT * DD);     // [B,k]
    int*   ws_a    = ws_e + BT * KK;                        // [B,k]
    float* ws_w    = (float*)(ws_a + BT * KK);              // [B,k]

    // GEMM1: h_pre = silu(z @ fc1_w^T + fc1_b)
    wmma_gemm_bias_act<<<dim3(HH / 64, BT / 128), 256, 0, stream>>>(
        z, fc1_w, fc1_b, h_pre, BT, HH, DD, 1);

    // GEMM2: out_pre = h_pre @ fc2_w^T + fc2_b
    wmma_gemm_bias_act<<<dim3(DD / 64, BT / 128), 256, 0, stream>>>(
        h_pre, fc2_w, fc2_b, out_pre, BT, DD, HH, 0);

    // router probabilities (+ zero used_mask)
    router_probs<<<BT / 8, 256, 0, stream>>>(
        z, proto, ebias, tau, eps, ban, out + O_PROBS, out + O_MASK);

    // sequential capacity routing (LDS-staged, wave-cooperative)
    routing_scan<<<1, 256, 0, stream>>>(
        out + O_PROBS, top_k, cap,
        out + O_AIDX, out + O_AW, out + O_MASK, out + O_EVE, out + O_HIT,
        ws_e, ws_w);

    // adapter selection
    adapter_argmax<<<(BT * KK) / 8, 256, 0, stream>>>(
        z, akeys, ws_e, ws_a, out + O_EVA);

    // LoRA + combine -> out[:B*D]
    lora_combine<<<BT, 256, 0, stream>>>(
        z, h_pre, out_pre, a1, b1, a2, b2, ws_e, ws_a, ws_w, out + O_Z);
}